// Attn_38757784879237
// MI455X (gfx1250) — compile-verified
//
#include <hip/hip_runtime.h>
#include <hip/hip_bf16.h>

// ---------------------------------------------------------------------------
// Sliding-window self-attention smoothing + Linear(2R,R), fp32 WMMA (gfx1250)
//   L = 32768, R = 128, WINDOW = 33 (pad 16)
// One wave32 per 16-row tile; 6 tiles per workgroup. All three GEMM stages
// run on V_WMMA_F32_16X16X4_F32. The Linear weights (w1, 128KB) are staged
// transposed into block-shared LDS once (CDNA5: 320KB LDS per WGP), so the
// dominant GEMM (K=256) feeds both operands from conflict-free LDS.
// ---------------------------------------------------------------------------

#define RANK    128
#define WIN     33
#define HALF    16          // (WIN-1)/2
#define TROWS   48          // 16 tile rows + 16 halo each side
#define TILE_M  16
#define WAVES   6           // waves (tiles) per workgroup

// LDS strides (floats), padded for bank-conflict-free column access
#define TSTRIDE 132         // 48 x 132 : staged tf rows (zero padded halo)
#define ASTRIDE 52          // 16 x 52  : scores -> attention weights
#define SLAB_F  (TROWS * TSTRIDE + TILE_M * ASTRIDE)   // result aliases rows 0..15
#define W1S     137         // w1T row stride (odd -> conflict-free frag reads)
#define W1_F    (2 * RANK * W1S)

typedef float v2f __attribute__((ext_vector_type(2)));
typedef float v4f __attribute__((ext_vector_type(4)));
typedef float v8f __attribute__((ext_vector_type(8)));

__device__ __forceinline__ v8f wmma_f32(v2f a, v2f b, v8f c) {
    // V_WMMA_F32_16X16X4_F32 : D(16x16,f32) = A(16x4,f32) * B(4x16,f32) + C
    return __builtin_amdgcn_wmma_f32_16x16x4_f32(
        /*neg_a=*/false, a, /*neg_b=*/false, b,
        /*c_mod=*/(short)0, c, /*reuse_a=*/false, /*reuse_b=*/false);
}

extern "C" __global__ void attn_smooth_kernel(
    const float* __restrict__ tf,   // [L, 128]
    const float* __restrict__ w1,   // [128, 256]  (out, in) row-major
    const float* __restrict__ b1,   // [128]
    float* __restrict__ out,        // [L, 128]
    int L, int ntiles)
{
    extern __shared__ float lds[];

    const int lane  = threadIdx.x & 31;
    const int wv    = threadIdx.x >> 5;
    int tile = blockIdx.x * WAVES + wv;
    if (tile >= ntiles) tile = ntiles - 1;      // tail waves: redo last tile
    const int t0    = tile * TILE_M;

    const int mrow  = lane & 15;    // M (A/C row) or N (B/C col) index
    const int khalf = lane >> 4;    // K sub-phase for A/B fragments

    float* tileS = lds + wv * SLAB_F;             // [48][TSTRIDE]
    float* attnS = tileS + TROWS * TSTRIDE;       // [16][ASTRIDE]
    float* resS  = tileS;                         // rows 0..15, dead after C
    float* w1T   = lds + WAVES * SLAB_F;          // [256][W1S] block-shared

    // ---- Stage w1 transposed into block-shared LDS (coalesced reads) -----
    // chunk c covers w1 flat floats [4c, 4c+3]; n = row, k = col of w1[n][k]
    for (int c = threadIdx.x; c < (2 * RANK * RANK) / 4; c += WAVES * 32) {
        int n = c >> 6;
        int k = (c & 63) * 4;
        v4f v = *(const v4f*)(w1 + (size_t)n * 256 + k);
        w1T[(k + 0) * W1S + n] = v.x;
        w1T[(k + 1) * W1S + n] = v.y;
        w1T[(k + 2) * W1S + n] = v.z;
        w1T[(k + 3) * W1S + n] = v.w;
    }

    // ---- Stage 48 rows (zero-padded halo) of time_factor into LDS --------
    for (int r = 0; r < TROWS; ++r) {
        int g = t0 - HALF + r;
        v4f v = {0.f, 0.f, 0.f, 0.f};
        if (g >= 0 && g < L)
            v = *(const v4f*)(tf + (size_t)g * RANK + lane * 4);
        *(v4f*)(tileS + r * TSTRIDE + lane * 4) = v;
    }
    __syncthreads();

    // A-fragment base for rows tf[t0 + m] (phases A and D first half)
    const float* aBase = tileS + (HALF + mrow) * TSTRIDE + 2 * khalf;

    // ---- Phase A: scores(16x48) = tf_tile(16x128) x tf_halo^T(128x48) ----
    // Zero-padded halo rows naturally yield score == 0, matching the
    // reference's zero-padding semantics (they stay in the softmax).
    {
        v8f acc0 = {}, acc1 = {}, acc2 = {};
        #pragma unroll 4
        for (int k = 0; k < RANK / 4; ++k) {
            v2f a;
            a.x = aBase[4 * k];
            a.y = aBase[4 * k + 1];
            const float* bb = tileS + mrow * TSTRIDE + 4 * k + 2 * khalf;
            v2f b0, b1v, b2;
            b0.x  = bb[0];                 b0.y  = bb[1];
            b1v.x = bb[16 * TSTRIDE];      b1v.y = bb[16 * TSTRIDE + 1];
            b2.x  = bb[32 * TSTRIDE];      b2.y  = bb[32 * TSTRIDE + 1];
            acc0 = wmma_f32(a, b0,  acc0);
            acc1 = wmma_f32(a, b1v, acc1);
            acc2 = wmma_f32(a, b2,  acc2);
        }
        #pragma unroll
        for (int v = 0; v < 8; ++v) {
            int m = v + 8 * khalf;
            attnS[m * ASTRIDE +  0 + mrow] = acc0[v];
            attnS[m * ASTRIDE + 16 + mrow] = acc1[v];
            attnS[m * ASTRIDE + 32 + mrow] = acc2[v];
        }
    }
    __syncthreads();

    // ---- Phase B: softmax over the 33 window positions j in [m, m+32] ----
    if (lane < 16) {
        const int m = lane;
        float* srow = attnS + m * ASTRIDE;
        float mx = -3.4e38f;
        #pragma unroll
        for (int j = 0; j < WIN; ++j) mx = fmaxf(mx, srow[m + j]);
        float sum = 0.f;
        #pragma unroll
        for (int j = 0; j < WIN; ++j) {
            float e = __expf(srow[m + j] - mx);
            srow[m + j] = e;
            sum += e;
        }
        float inv = 1.f / sum;
        // zero outside-window slots, normalize inside
        #pragma unroll
        for (int j = 0; j < TROWS; ++j) {
            float v = (j >= m && j <= m + 32) ? srow[j] * inv : 0.f;
            srow[j] = v;
        }
    }
    __syncthreads();

    // ---- Phase C: result(16x128) = attn(16x48) x tf_halo(48x128) ---------
    // Accumulate fully in registers, then store into rows 0..15 of tileS
    // (the low halo is dead after this phase's reads; DS ops are in-order).
    {
        v8f acc[8] = {};
        #pragma unroll 3
        for (int k = 0; k < TROWS / 4; ++k) {
            v2f a;
            a.x = attnS[mrow * ASTRIDE + 4 * k + 2 * khalf];
            a.y = attnS[mrow * ASTRIDE + 4 * k + 2 * khalf + 1];
            const float* bb = tileS + (4 * k + 2 * khalf) * TSTRIDE + mrow;
            #pragma unroll
            for (int nb = 0; nb < 8; ++nb) {
                v2f b;
                b.x = bb[16 * nb];
                b.y = bb[16 * nb + TSTRIDE];
                acc[nb] = wmma_f32(a, b, acc[nb]);
            }
        }
        #pragma unroll
        for (int nb = 0; nb < 8; ++nb)
            #pragma unroll
            for (int v = 0; v < 8; ++v)
                resS[(v + 8 * khalf) * TSTRIDE + 16 * nb + mrow] = acc[nb][v];
    }
    __syncthreads();

    // ---- Phase D: out(16x128) = [tf | result](16x256) x w1T + b1 ---------
    // Both operands from LDS: A frags from the tile slab, B frags from the
    // block-shared transposed weights.
    {
        v8f acc[8] = {};
        // K = 0..127 : tf tile rows
        #pragma unroll 2
        for (int k = 0; k < RANK / 4; ++k) {
            v2f a;
            a.x = aBase[4 * k];
            a.y = aBase[4 * k + 1];
            const float* wb = w1T + (4 * k + 2 * khalf) * W1S + mrow;
            #pragma unroll
            for (int nb = 0; nb < 8; ++nb) {
                v2f b;                      // B[k][n] = w1T[k][n]
                b.x = wb[16 * nb];
                b.y = wb[16 * nb + W1S];
                acc[nb] = wmma_f32(a, b, acc[nb]);
            }
        }
        // K = 128..255 : smoothed result rows
        #pragma unroll 2
        for (int k = 0; k < RANK / 4; ++k) {
            v2f a;
            a.x = resS[mrow * TSTRIDE + 4 * k + 2 * khalf];
            a.y = resS[mrow * TSTRIDE + 4 * k + 2 * khalf + 1];
            const float* wb = w1T + (RANK + 4 * k + 2 * khalf) * W1S + mrow;
            #pragma unroll
            for (int nb = 0; nb < 8; ++nb) {
                v2f b;
                b.x = wb[16 * nb];
                b.y = wb[16 * nb + W1S];
                acc[nb] = wmma_f32(a, b, acc[nb]);
            }
        }
        #pragma unroll
        for (int nb = 0; nb < 8; ++nb) {
            int n = 16 * nb + mrow;
            float bias = b1[n];
            #pragma unroll
            for (int v = 0; v < 8; ++v) {
                int m = v + 8 * khalf;
                out[(size_t)(t0 + m) * RANK + n] = acc[nb][v] + bias;
            }
        }
    }
}

extern "C" void kernel_launch(void* const* d_in, const int* in_sizes, int n_in,
                              void* d_out, int out_size, void* d_ws, size_t ws_size,
                              hipStream_t stream) {
    const float* tf  = (const float*)d_in[0];   // [L,128]
    const float* w1  = (const float*)d_in[1];   // [128,256]
    const float* b1  = (const float*)d_in[2];   // [128]
    float* out = (float*)d_out;

    const int L      = in_sizes[0] / RANK;          // 32768
    const int tiles  = L / TILE_M;                  // 2048
    const int blocks = (tiles + WAVES - 1) / WAVES; // 342
    const size_t shmem = (size_t)(WAVES * SLAB_F + W1_F) * sizeof(float);

    attn_smooth_kernel<<<blocks, 32 * WAVES, shmem, stream>>>(tf, w1, b1, out,
                                                              L, tiles);
}